// GNNModel_68779606278426
// MI455X (gfx1250) — compile-verified
//
#include <hip/hip_runtime.h>
#include <math.h>

typedef __attribute__((ext_vector_type(2))) float v2f;
typedef __attribute__((ext_vector_type(8))) float v8f;

// ---------------------------------------------------------------------------
// Small elementwise helpers
// ---------------------------------------------------------------------------
__global__ __launch_bounds__(256) void k_fill(float* __restrict__ p, float v, int n) {
  int i = blockIdx.x * 256 + threadIdx.x;
  if (i < n) p[i] = v;
}

__global__ __launch_bounds__(256) void k_degcount(const int* __restrict__ dst,
                                                  float* __restrict__ deg, int E) {
  int i = blockIdx.x * 256 + threadIdx.x;
  if (i < E) atomicAdd(&deg[dst[i]], 1.0f);
}

__global__ __launch_bounds__(256) void k_rsqrt_inplace(float* __restrict__ p, int n) {
  int i = blockIdx.x * 256 + threadIdx.x;
  if (i < n) {
    float d = p[i];
    p[i] = (d > 0.0f) ? rsqrtf(d) : 0.0f;
  }
}

// ---------------------------------------------------------------------------
// fp32 WMMA GEMM: C[M,N] = A[M,K] * B[K,N].
// One wave32 per 16-row stripe: computes all N/16 tiles of that stripe so each
// A fragment is loaded once and reused across N (8x less A traffic than
// tile-per-wave). B is tiny (<=64KB) and stays resident in WGP$/L0.
// A 16x4 fragment: lane L holds row (L&15); VGPR0/1 = K {0,1} (lanes 0-15) or
// K {2,3} (lanes 16-31). B 4x16 mirrored (lane = column). C/D: VGPR r -> row
// (half*8 + r), col = lane&15. (ISA 7.12.2)  M%16==0, N%16==0, K%4==0.
// ---------------------------------------------------------------------------
template <int K, int N>
__global__ __launch_bounds__(256) void k_gemm_wmma_f32(const float* __restrict__ A,
                                                       const float* __restrict__ B,
                                                       float* __restrict__ C, int M) {
  constexpr int NT = N / 16;                 // N tiles per row stripe
  int stripe = blockIdx.x * 8 + (threadIdx.x >> 5);
  if (stripe >= M / 16) return;              // uniform per wave
  int lane = threadIdx.x & 31;
  int half = lane >> 4;                      // 0: K pair {0,1}, 1: K pair {2,3}
  int l16  = lane & 15;

  const float* arow  = A + (size_t)(stripe * 16 + l16) * K + half * 2;
  const float* bbase = B + l16;

  v8f acc[NT];
#pragma unroll
  for (int t = 0; t < NT; ++t) acc[t] = (v8f){};

#pragma unroll 2
  for (int k = 0; k < K; k += 4) {
    float2 av = *(const float2*)(arow + k);  // contiguous K pair, 8B aligned
    v2f a; a.x = av.x; a.y = av.y;
#pragma unroll
    for (int t = 0; t < NT; ++t) {
      const float* bp = bbase + t * 16;
      v2f b;
      b.x = bp[(size_t)(k + half * 2 + 0) * N];
      b.y = bp[(size_t)(k + half * 2 + 1) * N];
      acc[t] = __builtin_amdgcn_wmma_f32_16x16x4_f32(
          /*neg_a=*/false, a, /*neg_b=*/false, b,
          /*c_mod=*/(short)0, acc[t], /*reuse_a=*/false, /*reuse_b=*/false);
    }
  }

  float* crow = C + (size_t)(stripe * 16 + half * 8) * N + l16;
#pragma unroll
  for (int t = 0; t < NT; ++t) {
#pragma unroll
    for (int r = 0; r < 8; ++r) crow[(size_t)r * N + t * 16] = acc[t][r];
  }
}

// ---------------------------------------------------------------------------
// Self-loop term: out[v,f] = xw[v,f] * dinv[v]^2
// ---------------------------------------------------------------------------
template <int F>
__global__ __launch_bounds__(256) void k_selfloop(const float* __restrict__ xw,
                                                  const float* __restrict__ dinv,
                                                  float* __restrict__ out, int nNodes) {
  int i = blockIdx.x * 256 + threadIdx.x;
  if (i >= nNodes * F) return;
  int v = i / F;
  float dv = dinv[v];
  out[i] = xw[i] * (dv * dv);
}

// ---------------------------------------------------------------------------
// Edge scatter: one wave per edge; out[dst] += xw[src] * dinv[s]*dinv[d]
// Gather rows live in L2 (25.6 MB << 192 MB); atomics resolve at L2.
// ---------------------------------------------------------------------------
template <int F>
__global__ __launch_bounds__(256) void k_scatter(const int* __restrict__ src,
                                                 const int* __restrict__ dst,
                                                 const float* __restrict__ dinv,
                                                 const float* __restrict__ xw,
                                                 float* __restrict__ out, int E) {
  int e = blockIdx.x * 8 + (threadIdx.x >> 5);
  if (e >= E) return;
  int lane = threadIdx.x & 31;
  int s = src[e];
  int d = dst[e];
  float norm = dinv[s] * dinv[d];
  const float* xs = xw + (size_t)s * F;
  float* od = out + (size_t)d * F;
#pragma unroll
  for (int f = lane; f < F; f += 32) atomicAdd(&od[f], xs[f] * norm);
}

// ---------------------------------------------------------------------------
// h = relu(h + bias)
// ---------------------------------------------------------------------------
template <int F>
__global__ __launch_bounds__(256) void k_bias_relu(float* __restrict__ h,
                                                   const float* __restrict__ b, int nNodes) {
  int i = blockIdx.x * 256 + threadIdx.x;
  if (i >= nNodes * F) return;
  h[i] = fmaxf(h[i] + b[i % F], 0.0f);
}

// ---------------------------------------------------------------------------
// In-place row log_softmax over 64 classes (+bias): one wave32 per row,
// 2 columns per lane, shuffle reductions (wave32 widths only).
// ---------------------------------------------------------------------------
__global__ __launch_bounds__(256) void k_bias_logsoftmax64(float* __restrict__ out,
                                                           const float* __restrict__ b,
                                                           int nRows) {
  int row = blockIdx.x * 8 + (threadIdx.x >> 5);
  if (row >= nRows) return;
  int lane = threadIdx.x & 31;
  float* r = out + (size_t)row * 64;
  float v0 = r[lane]      + b[lane];
  float v1 = r[lane + 32] + b[lane + 32];
  float m = fmaxf(v0, v1);
#pragma unroll
  for (int o = 16; o > 0; o >>= 1) m = fmaxf(m, __shfl_xor(m, o, 32));
  float s = expf(v0 - m) + expf(v1 - m);
#pragma unroll
  for (int o = 16; o > 0; o >>= 1) s += __shfl_xor(s, o, 32);
  float ls = m + logf(s);
  r[lane]      = v0 - ls;
  r[lane + 32] = v1 - ls;
}

// ---------------------------------------------------------------------------
// Launch: x, edge_index, W1, b1, W2, b2  ->  log_softmax(GCN2(relu(GCN1(x))))
// ---------------------------------------------------------------------------
extern "C" void kernel_launch(void* const* d_in, const int* in_sizes, int n_in,
                              void* d_out, int out_size, void* d_ws, size_t ws_size,
                              hipStream_t stream) {
  const float* x  = (const float*)d_in[0];
  const int*   ei = (const int*)d_in[1];
  const float* W1 = (const float*)d_in[2];
  const float* b1 = (const float*)d_in[3];
  const float* W2 = (const float*)d_in[4];
  const float* b2 = (const float*)d_in[5];

  const int F = 128, H = 128, C = 64;
  const int nNodes = in_sizes[0] / F;   // 50000 (multiple of 16)
  const int E      = in_sizes[1] / 2;   // 800000
  const int* src = ei;
  const int* dst = ei + E;

  // workspace layout (floats): deg/dinv | xw1 | h1 | xw2
  float* ws = (float*)d_ws;
  size_t o = 0;
  float* deg = ws + o; o += ((size_t)nNodes + 255) & ~(size_t)255;
  float* xw1 = ws + o; o += (size_t)nNodes * H;
  float* h1  = ws + o; o += (size_t)nNodes * H;
  float* xw2 = ws + o; o += (size_t)nNodes * C;
  float* out = (float*)d_out;

  auto cdiv = [](long long a, long long b) { return (int)((a + b - 1) / b); };

  // --- symmetric normalization: dinv = rsqrt(1 + indeg) ---
  k_fill<<<cdiv(nNodes, 256), 256, 0, stream>>>(deg, 1.0f, nNodes);
  k_degcount<<<cdiv(E, 256), 256, 0, stream>>>(dst, deg, E);
  k_rsqrt_inplace<<<cdiv(nNodes, 256), 256, 0, stream>>>(deg, nNodes); // deg := dinv

  const int stripes = nNodes / 16;  // 3125 row stripes

  // --- layer 1: xw1 = x @ W1 ; h1 = A_hat @ xw1 + b1 ; relu ---
  k_gemm_wmma_f32<128, 128><<<cdiv(stripes, 8), 256, 0, stream>>>(x, W1, xw1, nNodes);
  k_selfloop<128><<<cdiv((long long)nNodes * H, 256), 256, 0, stream>>>(xw1, deg, h1, nNodes);
  k_scatter<128><<<cdiv(E, 8), 256, 0, stream>>>(src, dst, deg, xw1, h1, E);
  k_bias_relu<128><<<cdiv((long long)nNodes * H, 256), 256, 0, stream>>>(h1, b1, nNodes);

  // --- layer 2: xw2 = h1 @ W2 ; out = A_hat @ xw2 + b2 ; log_softmax ---
  k_gemm_wmma_f32<128, 64><<<cdiv(stripes, 8), 256, 0, stream>>>(h1, W2, xw2, nNodes);
  k_selfloop<64><<<cdiv((long long)nNodes * C, 256), 256, 0, stream>>>(xw2, deg, out, nNodes);
  k_scatter<64><<<cdiv(E, 8), 256, 0, stream>>>(src, dst, deg, xw2, out, E);
  k_bias_logsoftmax64<<<cdiv(nNodes, 8), 256, 0, stream>>>(out, b2, nNodes);
}